// GCNEncoder_24386824307023
// MI455X (gfx1250) — compile-verified
//
#include <hip/hip_runtime.h>
#include <hip/hip_bf16.h>

typedef float v2f __attribute__((ext_vector_type(2)));
typedef float v8f __attribute__((ext_vector_type(8)));

#define N_GRAPHS 64

// ---------------- utility kernels ----------------
__global__ void k_zero_i32(int* p, int n) {
  int i = blockIdx.x * blockDim.x + threadIdx.x;
  for (; i < n; i += gridDim.x * blockDim.x) p[i] = 0;
}
__global__ void k_zero_f32(float* p, int n) {
  int i = blockIdx.x * blockDim.x + threadIdx.x;
  for (; i < n; i += gridDim.x * blockDim.x) p[i] = 0.0f;
}

// ---------------- CSR build ----------------
__global__ void k_count(const int* __restrict__ dst, int* __restrict__ cnt, int E) {
  int e = blockIdx.x * blockDim.x + threadIdx.x;
  if (e < E) atomicAdd(&cnt[dst[e]], 1);
}

// single-block exclusive scan of cnt -> rowstart, plus dinv = rsqrt(cnt+1)
__global__ void k_scan(const int* __restrict__ cnt, int* __restrict__ rowstart,
                       float* __restrict__ dinv, int n) {
  __shared__ int tmp[1024];
  __shared__ int carry;
  const int tid = threadIdx.x;
  if (tid == 0) carry = 0;
  __syncthreads();
  for (int base = 0; base < n; base += 1024) {
    int i = base + tid;
    int v = (i < n) ? cnt[i] : 0;
    tmp[tid] = v;
    __syncthreads();
    #pragma unroll
    for (int off = 1; off < 1024; off <<= 1) {
      int t = (tid >= off) ? tmp[tid - off] : 0;
      __syncthreads();
      tmp[tid] += t;
      __syncthreads();
    }
    int incl = tmp[tid];
    int c = carry;
    if (i < n) {
      rowstart[i] = c + incl - v;
      dinv[i] = rsqrtf((float)v + 1.0f);
    }
    __syncthreads();
    if (tid == 0) carry = c + tmp[1023];
    __syncthreads();
  }
  if (tid == 0) rowstart[n] = carry;
}

__global__ void k_fill(const int* __restrict__ src, const int* __restrict__ dst,
                       const int* __restrict__ rowstart, int* __restrict__ cur,
                       int* __restrict__ csr, int E) {
  int e = blockIdx.x * blockDim.x + threadIdx.x;
  if (e < E) {
    int d = dst[e];
    int pos = atomicAdd(&cur[d], 1);
    csr[rowstart[d] + pos] = src[e];
  }
}

// ---------------- f32 WMMA GEMM: out[M,N] = (A[M,128] * W[128,N]) * dinv[row] ----------------
// Block = 256 threads = 8 waves, one 16-row strip per wave.
// W is staged in LDS in a k-pair-interleaved layout: Wp[k/2][col][k%2], so each
// B fragment (b.x = W[ka][col], b.y = W[ka+1][col]) is ONE aligned ds_load_b64
// directly into a consecutive VGPR pair -- no pack movs in the WMMA loop.
// Pair-row stride = 2N+32 dwords => the two lane-halves (k-pairs kb/2 and
// kb/2+1) land on disjoint 32-bank groups: a wave covers all 64 banks.
template <int N>
__global__ void k_gemm_wmma(const float* __restrict__ A, const float* __restrict__ W,
                            const float* __restrict__ dinv, float* __restrict__ out, int M) {
  constexpr int K = 128;
  constexpr int NT = N / 16;
  constexpr int RP = 2 * N + 32;   // dwords per k-pair row (== 32 mod 64)
  extern __shared__ float Wlds[];  // (K/2)*RP floats
  for (int i = threadIdx.x; i < K * N; i += blockDim.x) {
    int k = i / N;
    int n = i % N;
    Wlds[(k >> 1) * RP + n * 2 + (k & 1)] = W[i];
  }
  __syncthreads();

  const int wave = threadIdx.x >> 5;
  const int lane = threadIdx.x & 31;
  const int mtile = blockIdx.x * 8 + wave;
  if (mtile * 16 >= M) return;  // wave-uniform: EXEC stays all-ones for WMMA

  const int Rm = mtile * 16;
  const int half = lane >> 4;   // 0: lanes hold K0/K1, 1: lanes hold K2/K3
  const int l16 = lane & 15;

  v8f acc[NT] = {};
  const float* arow = A + (size_t)(Rm + l16) * K;
  // per-lane base into LDS: k-pair (half) and column lane offset
  const float* bbase = Wlds + half * RP + l16 * 2;

  for (int kb = 0; kb < K; kb += 4) {
    // A fragment (16x4 f32): lane<16 -> A[m][kb..kb+1], lane>=16 -> A[m][kb+2..kb+3]
    v2f a = *(const v2f*)(arow + kb + 2 * half);
    const float* bk = bbase + (kb >> 1) * RP;
    #pragma unroll
    for (int t = 0; t < NT; ++t) {
      // B fragment (4x16 f32): single b64 load of {W[ka][col], W[ka+1][col]}
      v2f b = *(const v2f*)(bk + t * 32);
      acc[t] = __builtin_amdgcn_wmma_f32_16x16x4_f32(
          false, a, false, b, (short)0, acc[t], false, false);
    }
  }

  // C/D layout: vgpr r, lanes 0-15 -> row Rm+r, lanes 16-31 -> row Rm+8+r.
  // Epilogue fuses the D^-1/2 row scaling: store hws = (A@W) * dinv[row].
  #pragma unroll
  for (int r = 0; r < 8; ++r) {
    const int row = Rm + r + 8 * half;
    const float di = dinv[row];
    #pragma unroll
    for (int t = 0; t < NT; ++t) {
      out[(size_t)row * N + t * 16 + l16] = acc[t][r] * di;
    }
  }
}

// ---------------- gather-side aggregation on pre-scaled features ----------------
// hws = (h@W) * dinv  =>  out = relu(dinv[i] * (hws[i] + sum_{e:dst=i} hws[src_e]) + b)
template <int F>
__global__ void k_aggregate(const float* __restrict__ hws, const float* __restrict__ dinv,
                            const int* __restrict__ rowstart, const int* __restrict__ csr,
                            const float* __restrict__ bias, float* __restrict__ out, int n) {
  constexpr int NPB = 256 / F;  // nodes per 256-thread block
  const int node = blockIdx.x * NPB + threadIdx.x / F;
  const int f = threadIdx.x % F;
  if (node >= n) return;
  const int s = rowstart[node];
  const int e = rowstart[node + 1];
  float sum = hws[(size_t)node * F + f];  // self-loop term (already * dinv[node])
  for (int j = s; j < e; ++j) {
    sum += hws[(size_t)csr[j] * F + f];
  }
  float v = dinv[node] * sum + bias[f];
  out[(size_t)node * F + f] = fmaxf(v, 0.0f);
}

// ---------------- mean pool ----------------
__global__ void k_pool_acc(const float* __restrict__ h, const int* __restrict__ batch,
                           float* __restrict__ pool, float* __restrict__ cntg, int n) {
  int idx = blockIdx.x * blockDim.x + threadIdx.x;
  if (idx < n * 64) {
    int node = idx >> 6;
    int f = idx & 63;
    int g = batch[node];
    atomicAdd(&pool[g * 64 + f], h[idx]);
    if (f == 0) atomicAdd(&cntg[g], 1.0f);
  }
}

__global__ void k_pool_final(const float* __restrict__ pool, const float* __restrict__ cntg,
                             float* __restrict__ out) {
  int idx = blockIdx.x * blockDim.x + threadIdx.x;
  if (idx < N_GRAPHS * 64) {
    out[idx] = pool[idx] / fmaxf(cntg[idx >> 6], 1.0f);
  }
}

// ---------------- launcher ----------------
extern "C" void kernel_launch(void* const* d_in, const int* in_sizes, int n_in,
                              void* d_out, int out_size, void* d_ws, size_t ws_size,
                              hipStream_t stream) {
  const float* x     = (const float*)d_in[0];
  const float* W1    = (const float*)d_in[1];
  const float* b1    = (const float*)d_in[2];
  const float* W2    = (const float*)d_in[3];
  const float* b2    = (const float*)d_in[4];
  const int*   ei    = (const int*)d_in[5];
  const int*   batch = (const int*)d_in[6];

  const int N = in_sizes[6];      // 50000 nodes
  const int E = in_sizes[5] / 2;  // 800000 edges
  const int G = N_GRAPHS;

  const int* src = ei;
  const int* dst = ei + E;

  // workspace layout (256B aligned)
  char* base = (char*)d_ws;
  size_t off = 0;
  auto take = [&](size_t bytes) {
    size_t o = off;
    off = (off + bytes + 255) & ~(size_t)255;
    return o;
  };
  int*   cnt      = (int*)(base + take(sizeof(int) * (size_t)N));
  int*   cur      = (int*)(base + take(sizeof(int) * (size_t)N));
  int*   rowstart = (int*)(base + take(sizeof(int) * (size_t)(N + 1)));
  float* dinv     = (float*)(base + take(sizeof(float) * (size_t)N));
  int*   csr      = (int*)(base + take(sizeof(int) * (size_t)E));
  float* hw1      = (float*)(base + take(sizeof(float) * (size_t)N * 128));  // reused as hw2
  float* h1       = (float*)(base + take(sizeof(float) * (size_t)N * 128));  // reused as h2
  float* pool     = (float*)(base + take(sizeof(float) * (size_t)G * 64));
  float* cntg     = (float*)(base + take(sizeof(float) * (size_t)G));
  float* hw2 = hw1;
  float* h2  = h1;

  const int B = 256;
  const int gN = (N + B - 1) / B;
  const int gE = (E + B - 1) / B;
  const int mtiles = (N + 15) / 16;
  const int gGemm = (mtiles + 7) / 8;

  // 1) degree histogram + scan + CSR fill (dinv needed before the GEMMs)
  k_zero_i32<<<gN, B, 0, stream>>>(cnt, N);
  k_count<<<gE, B, 0, stream>>>(dst, cnt, E);
  k_scan<<<1, 1024, 0, stream>>>(cnt, rowstart, dinv, N);
  k_zero_i32<<<gN, B, 0, stream>>>(cur, N);
  k_fill<<<gE, B, 0, stream>>>(src, dst, rowstart, cur, csr, E);

  // 2) layer 1: hws1 = (x @ W1) * dinv (WMMA), then aggregation + bias + relu
  k_gemm_wmma<128><<<gGemm, 256, 64 * (2 * 128 + 32) * sizeof(float), stream>>>(x, W1, dinv, hw1, N);
  k_aggregate<128><<<(N + 1) / 2, 256, 0, stream>>>(hw1, dinv, rowstart, csr, b1, h1, N);

  // 3) layer 2: hws2 = (h1 @ W2) * dinv (WMMA), aggregation + bias + relu
  k_gemm_wmma<64><<<gGemm, 256, 64 * (2 * 64 + 32) * sizeof(float), stream>>>(h1, W2, dinv, hw2, N);
  k_aggregate<64><<<(N + 3) / 4, 256, 0, stream>>>(hw2, dinv, rowstart, csr, b2, h2, N);

  // 4) global mean pool
  k_zero_f32<<<(G * 64 + G + B - 1) / B, B, 0, stream>>>(pool, G * 64 + G);  // pool+cntg contiguous
  k_pool_acc<<<((size_t)N * 64 + B - 1) / B, B, 0, stream>>>(h2, batch, pool, cntg, N);
  k_pool_final<<<(G * 64 + B - 1) / B, B, 0, stream>>>(pool, cntg, (float*)d_out);
}